// GATModel_45320494907955
// MI455X (gfx1250) — compile-verified
//
#include <hip/hip_runtime.h>
#include <math.h>

// ---- problem constants (from reference) ----
#define NN   50000
#define FF   128
#define EE   800000
#define HIDC 32
#define NHEAD 8
#define NTGT 10
#define NGRP 64
#define NEG_SLOPE 0.2f
#define MT   5       // row tiles per wave in GEMM (3125 row tiles = 625 * 5)

typedef float v2f __attribute__((ext_vector_type(2)));
typedef float v8f __attribute__((ext_vector_type(8)));

// ---------------------------------------------------------------------------
// WMMA f32 GEMM: C[M,Nc] = A[M,K] @ B[K,Nc]; one wave computes MT 16x16 tiles
// stacked in M, sharing each B fragment across MT V_WMMA_F32_16X16X4_F32 ops.
// K loop is software-pipelined (register double buffering): next step's
// fragments are in flight while current step's WMMAs execute.
// ISA 7.12.2 lane layouts:
//   A 16x4:  lanes 0-15 -> M=lane, VGPR0=K0/VGPR1=K1; lanes 16-31 -> K2/K3
//   B 4x16:  lanes 0-15 -> N=lane, VGPR0=K0/VGPR1=K1; lanes 16-31 -> K2/K3
//   C 16x16: VGPR r: lanes 0-15 -> M=r, N=lane; lanes 16-31 -> M=r+8
// M multiple of 16*MT (50000 = 625*80), K multiple of 4, Nc multiple of 16.
// ---------------------------------------------------------------------------
__global__ __launch_bounds__(32) void gemm_wmma_f32_kernel(
    const float* __restrict__ A, const float* __restrict__ B,
    float* __restrict__ C, int K, int Nc) {
  const int rowBase = blockIdx.x * (16 * MT);
  const int colBase = blockIdx.y * 16;
  const int lane = threadIdx.x;
  const int half = lane >> 4;   // 0: K0/K1 slice, 1: K2/K3 slice
  const int l15  = lane & 15;

  // per-lane base pointers; advanced by pure adds each K-step
  const float* ap[MT];
#pragma unroll
  for (int t = 0; t < MT; ++t)
    ap[t] = A + (size_t)(rowBase + t * 16 + l15) * K + half * 2;
  const float* bp = B + (size_t)(half * 2) * Nc + colBase + l15;

  v8f acc[MT] = {};

  // prologue: load step-0 fragments
  v2f bcur, acur[MT];
  bcur.x = bp[0];
  bcur.y = bp[Nc];
#pragma unroll
  for (int t = 0; t < MT; ++t) {
    acur[t].x = ap[t][0];
    acur[t].y = ap[t][1];
  }

  const int steps = K >> 2;
  for (int it = 0; it < steps - 1; ++it) {
    // prefetch next step's fragments (loads issue before the WMMAs below)
    bp += (size_t)4 * Nc;
    v2f bnxt;
    bnxt.x = bp[0];
    bnxt.y = bp[Nc];
    v2f anxt[MT];
#pragma unroll
    for (int t = 0; t < MT; ++t) {
      anxt[t].x = ap[t][4];
      anxt[t].y = ap[t][5];
      ap[t] += 4;
    }
    // consume current fragments
#pragma unroll
    for (int t = 0; t < MT; ++t) {
      acc[t] = __builtin_amdgcn_wmma_f32_16x16x4_f32(
          /*neg_a=*/false, acur[t], /*neg_b=*/false, bcur,
          /*c_mod=*/(short)0, acc[t], /*reuse_a=*/false, /*reuse_b=*/false);
    }
    bcur = bnxt;
#pragma unroll
    for (int t = 0; t < MT; ++t) acur[t] = anxt[t];
  }
  // epilogue: last step
#pragma unroll
  for (int t = 0; t < MT; ++t) {
    acc[t] = __builtin_amdgcn_wmma_f32_16x16x4_f32(
        false, acur[t], false, bcur, (short)0, acc[t], false, false);
  }

  const int outCol = colBase + l15;
#pragma unroll
  for (int t = 0; t < MT; ++t) {
    const int outRow = rowBase + t * 16 + half * 8;
#pragma unroll
    for (int r = 0; r < 8; ++r) {
      C[(size_t)(outRow + r) * Nc + outCol] = acc[t][r];
    }
  }
}

// ---------------------------------------------------------------------------
// Attention scores: s_src[n,h] = <h[n,h,:], a_src[h,:]>, same for dst.
// ---------------------------------------------------------------------------
__global__ void scores_kernel(const float* __restrict__ h,
                              const float* __restrict__ a_src,
                              const float* __restrict__ a_dst,
                              float* __restrict__ s_src,
                              float* __restrict__ s_dst, int H) {
  int i = blockIdx.x * blockDim.x + threadIdx.x;
  if (i >= NN * H) return;
  int n = i / H, hh = i % H;
  const float* hp = h + (size_t)n * H * HIDC + (size_t)hh * HIDC;
  const float* ap = a_src + (size_t)hh * HIDC;
  const float* dp = a_dst + (size_t)hh * HIDC;
  float ss = 0.f, sd = 0.f;
#pragma unroll
  for (int c = 0; c < HIDC; ++c) {
    float v = hp[c];
    ss += v * ap[c];
    sd += v * dp[c];
  }
  s_src[i] = ss;
  s_dst[i] = sd;
}

__global__ void fill_kernel(float* __restrict__ p, int n, float v) {
  int i = blockIdx.x * blockDim.x + threadIdx.x;
  if (i < n) p[i] = v;
}

__device__ __forceinline__ float lrelu(float x) {
  return x > 0.f ? x : NEG_SLOPE * x;
}

// float atomic max via monotone int/uint mapping; init value must be -inf.
__device__ __forceinline__ void atomicMaxF(float* addr, float v) {
  if (v >= 0.f)
    atomicMax((int*)addr, __float_as_int(v));
  else
    atomicMin((unsigned int*)addr, __float_as_uint(v));
}

__device__ __forceinline__ void edge_sd(const int* __restrict__ ei, int e,
                                        int& s, int& d) {
  if (e < EE) {
    s = ei[e];
    d = ei[EE + e];
  } else {          // self loops appended
    s = e - EE;
    d = e - EE;
  }
}

// pass 1: segment max of leaky_relu(s_src[src]+s_dst[dst]) over dst
__global__ void edge_max_kernel(const int* __restrict__ ei,
                                const float* __restrict__ s_src,
                                const float* __restrict__ s_dst,
                                float* __restrict__ m, int H) {
  int t = blockIdx.x * blockDim.x + threadIdx.x;
  int total = (EE + NN) * H;
  if (t >= total) return;
  int e = t / H, hh = t % H;
  int s, d;
  edge_sd(ei, e, s, d);
  float ev = lrelu(s_src[s * H + hh] + s_dst[d * H + hh]);
  atomicMaxF(&m[d * H + hh], ev);
}

// pass 2: z[dst] += exp(e - m[dst])
__global__ void edge_sum_kernel(const int* __restrict__ ei,
                                const float* __restrict__ s_src,
                                const float* __restrict__ s_dst,
                                const float* __restrict__ m,
                                float* __restrict__ z, int H) {
  int t = blockIdx.x * blockDim.x + threadIdx.x;
  int total = (EE + NN) * H;
  if (t >= total) return;
  int e = t / H, hh = t % H;
  int s, d;
  edge_sd(ei, e, s, d);
  int dh = d * H + hh;
  float ev = lrelu(s_src[s * H + hh] + s_dst[dh]);
  atomicAdd(&z[dh], expf(ev - m[dh]));
}

// pass 3: out[dst,h,c] += alpha * h[src,h,c]; one thread per (edge, channel)
// so gathers and scatter-atomics are contiguous per edge.
__global__ void edge_aggr_kernel(const int* __restrict__ ei,
                                 const float* __restrict__ s_src,
                                 const float* __restrict__ s_dst,
                                 const float* __restrict__ m,
                                 const float* __restrict__ z,
                                 const float* __restrict__ h,
                                 float* __restrict__ out, int H) {
  const int HC = H * HIDC;
  int t = blockIdx.x * blockDim.x + threadIdx.x;
  int total = (EE + NN) * HC;           // fits in int32 (max 217.6M)
  if (t >= total) return;
  int e = t / HC;
  int idx = t - e * HC;
  int hh = idx / HIDC;
  int s, d;
  edge_sd(ei, e, s, d);
  int dh = d * H + hh;
  float ev = lrelu(s_src[s * H + hh] + s_dst[dh]);
  float alpha = expf(ev - m[dh]) / z[dh];
  atomicAdd(&out[(size_t)d * HC + idx], h[(size_t)s * HC + idx] * alpha);
}

// bias + optional ELU
__global__ void finalize_kernel(float* __restrict__ out,
                                const float* __restrict__ bias, int n, int HC,
                                int do_elu) {
  int i = blockIdx.x * blockDim.x + threadIdx.x;
  if (i >= n) return;
  float v = out[i] + bias[i % HC];
  if (do_elu) v = v > 0.f ? v : expm1f(v);
  out[i] = v;
}

// global mean pool (accumulate) over graphs; one thread per (node, channel)
__global__ void pool_kernel(const float* __restrict__ h,
                            const int* __restrict__ batch,
                            float* __restrict__ sums, float* __restrict__ cnt) {
  int t = blockIdx.x * blockDim.x + threadIdx.x;
  if (t >= NN * HIDC) return;
  int n = t / HIDC, c = t - n * HIDC;
  int g = batch[n];
  atomicAdd(&sums[g * HIDC + c], h[(size_t)n * HIDC + c]);
  if (c == 0) atomicAdd(&cnt[g], 1.0f);
}

// head: out[g,t] = (sums[g,:]/max(cnt,1)) @ lw + lb
__global__ void head_kernel(const float* __restrict__ sums,
                            const float* __restrict__ cnt,
                            const float* __restrict__ lw,
                            const float* __restrict__ lb,
                            float* __restrict__ out) {
  int t = blockIdx.x * blockDim.x + threadIdx.x;
  if (t >= NGRP * NTGT) return;
  int g = t / NTGT, k = t % NTGT;
  float c = cnt[g];
  c = c > 1.0f ? c : 1.0f;
  float acc = lb[k];
#pragma unroll
  for (int ch = 0; ch < HIDC; ++ch) {
    acc += (sums[g * HIDC + ch] / c) * lw[ch * NTGT + k];
  }
  out[t] = acc;
}

static inline int cdiv(int a, int b) { return (a + b - 1) / b; }

extern "C" void kernel_launch(void* const* d_in, const int* in_sizes, int n_in,
                              void* d_out, int out_size, void* d_ws,
                              size_t ws_size, hipStream_t stream) {
  (void)in_sizes; (void)n_in; (void)out_size; (void)ws_size;
  const float* x   = (const float*)d_in[0];
  const int*   ei  = (const int*)d_in[1];
  const int*   bat = (const int*)d_in[2];
  const float* w0  = (const float*)d_in[3];
  const float* a0s = (const float*)d_in[4];
  const float* a0d = (const float*)d_in[5];
  const float* b0  = (const float*)d_in[6];
  const float* w1  = (const float*)d_in[7];
  const float* a1s = (const float*)d_in[8];
  const float* a1d = (const float*)d_in[9];
  const float* b1  = (const float*)d_in[10];
  const float* w2  = (const float*)d_in[11];
  const float* a2s = (const float*)d_in[12];
  const float* a2d = (const float*)d_in[13];
  const float* b2  = (const float*)d_in[14];
  const float* lw  = (const float*)d_in[15];
  const float* lb  = (const float*)d_in[16];
  float* out = (float*)d_out;

  // workspace layout (floats)
  float* bufA  = (float*)d_ws;                          // [N, 256] transformed
  float* bufB  = bufA + (size_t)NN * 256;               // [N, 256] aggregated
  float* s_src = bufB + (size_t)NN * 256;               // [N, 8]
  float* s_dst = s_src + (size_t)NN * NHEAD;            // [N, 8]
  float* mbuf  = s_dst + (size_t)NN * NHEAD;            // [N, 8]
  float* zbuf  = mbuf + (size_t)NN * NHEAD;             // [N, 8]
  float* sums  = zbuf + (size_t)NN * NHEAD;             // [NG, 32]
  float* cnt   = sums + NGRP * HIDC;                    // [NG]

  const int TB = 256;

  auto run_layer = [&](const float* in, int K, const float* W,
                       const float* as_, const float* ad_, const float* bias,
                       int H, float* ht, float* og, int do_elu) {
    const int HC = H * HIDC;
    // 1) transform: ht = in @ W   (WMMA f32, 5 row tiles per wave, pipelined)
    dim3 ggrid(NN / (16 * MT), HC / 16);
    gemm_wmma_f32_kernel<<<ggrid, 32, 0, stream>>>(in, W, ht, K, HC);
    // 2) attention scores
    const int NH = NN * H;
    scores_kernel<<<cdiv(NH, TB), TB, 0, stream>>>(ht, as_, ad_, s_src, s_dst, H);
    // 3) init segment buffers + output accumulator
    fill_kernel<<<cdiv(NH, TB), TB, 0, stream>>>(mbuf, NH, -INFINITY);
    fill_kernel<<<cdiv(NH, TB), TB, 0, stream>>>(zbuf, NH, 0.0f);
    fill_kernel<<<cdiv(NN * HC, TB), TB, 0, stream>>>(og, NN * HC, 0.0f);
    // 4) three-pass segment softmax + aggregation over edges (+self loops)
    const int TE = (EE + NN) * H;
    edge_max_kernel<<<cdiv(TE, TB), TB, 0, stream>>>(ei, s_src, s_dst, mbuf, H);
    edge_sum_kernel<<<cdiv(TE, TB), TB, 0, stream>>>(ei, s_src, s_dst, mbuf, zbuf, H);
    const int TA = (EE + NN) * HC;
    edge_aggr_kernel<<<cdiv(TA, TB), TB, 0, stream>>>(ei, s_src, s_dst, mbuf,
                                                      zbuf, ht, og, H);
    // 5) bias + activation
    finalize_kernel<<<cdiv(NN * HC, TB), TB, 0, stream>>>(og, bias, NN * HC, HC,
                                                          do_elu);
  };

  // layer 0: x[N,128] -> bufB[N,256], ELU
  run_layer(x, FF, w0, a0s, a0d, b0, NHEAD, bufA, bufB, 1);
  // layer 1: bufB[N,256] -> bufB[N,256] (ht in bufA), ELU
  run_layer(bufB, NHEAD * HIDC, w1, a1s, a1d, b1, NHEAD, bufA, bufB, 1);
  // layer 2: bufB[N,256] -> bufB[N,32], no activation
  run_layer(bufB, NHEAD * HIDC, w2, a2s, a2d, b2, 1, bufA, bufB, 0);

  // global mean pool + linear head
  fill_kernel<<<cdiv(NGRP * HIDC, TB), TB, 0, stream>>>(sums, NGRP * HIDC, 0.0f);
  fill_kernel<<<1, TB, 0, stream>>>(cnt, NGRP, 0.0f);
  pool_kernel<<<cdiv(NN * HIDC, TB), TB, 0, stream>>>(bufB, bat, sums, cnt);
  head_kernel<<<cdiv(NGRP * NTGT, TB), TB, 0, stream>>>(sums, cnt, lw, lb, out);
}